// EpsBallPoints_37812892074552
// MI455X (gfx1250) — compile-verified
//
#include <hip/hip_runtime.h>

typedef __attribute__((ext_vector_type(2))) float v2f;
typedef __attribute__((ext_vector_type(8))) float v8f;

#define BATCH 8
#define SPTS  2048
#define NPTS  8192
#define NS    64
#define R2    0.04f

// One wave32 owns a 16-sample tile; V_WMMA_F32_16X16X4_F32 computes 16x16
// dot-product tiles (K=3 padded to 4); ballot-based ordered compaction builds
// the first-64 index lists; uniform early exit when all 16 rows are full.
// 32 coords (two WMMA tiles) per loop iteration.
__global__ __launch_bounds__(256) void ball_query_wmma(
    const float* __restrict__ samples, const float* __restrict__ coord,
    int* __restrict__ out)
{
  const int lane = threadIdx.x & 31;
  const int wave = threadIdx.x >> 5;
  const int tile = blockIdx.x * 8 + wave;          // 1024 tiles total
  const int b  = tile >> 7;                        // 128 tiles per batch
  const int s0 = (tile & 127) << 4;                // 16 samples per tile

  const bool lo16 = lane < 16;
  const int  c16  = lane & 15;

  // ---- A matrix: 16x4 (M x K), rows = samples, K = {x,y,z,0} ----
  // Layout: VGPR0 = K0 (lanes 0-15) | K2 (lanes 16-31); VGPR1 = K1 | K3
  const float* sp = samples + ((size_t)b * SPTS + s0 + c16) * 3;
  const float sx = sp[0], sy = sp[1], sz = sp[2];
  const float sn = sx * sx + sy * sy + sz * sz;

  v2f A;
  A.x = lo16 ? sx : sz;
  A.y = lo16 ? sy : 0.0f;

  // Per-D-register threshold: lanes 0-15 -> row r, lanes 16-31 -> row r+8
  float thr[8];
#pragma unroll
  for (int r = 0; r < 8; ++r)
    thr[r] = R2 - __shfl(sn, lo16 ? r : (r + 8), 32);

  int cntA[8], cntB[8], fstA[8], fstB[8];
#pragma unroll
  for (int r = 0; r < 8; ++r) { cntA[r] = 0; cntB[r] = 0; fstA[r] = NPTS; fstB[r] = NPTS; }

  const float* cb = coord + (size_t)b * NPTS * 3;
  int* outBase = out + ((size_t)b * SPTS + s0) * NS;

  // Append coords [nb, nb+16) for D register dr against column norms cnv.
#define PROCESS(dr, cnv, nb)                                                   \
  do {                                                                         \
    const float t_ = fmaf((dr), -2.0f, (cnv));                                 \
    const bool cond_ = t_ <= thr[r];                                           \
    const unsigned mask_ = (unsigned)__ballot(cond_);                          \
    if (mask_) {                                                               \
      const unsigned lom_ = mask_ & 0xffffu;                                   \
      const unsigned him_ = mask_ >> 16;                                       \
      const unsigned mym_ = lo16 ? lom_ : him_;                                \
      const int base_ = lo16 ? cntA[r] : cntB[r];                              \
      const int pos_ = base_ + __popc(mym_ & ((1u << c16) - 1u));              \
      if (cond_ && pos_ < NS) {                                                \
        const int row_ = r + (lo16 ? 0 : 8);                                   \
        outBase[(size_t)row_ * NS + pos_] = (nb) + c16;                        \
      }                                                                        \
      if (cntA[r] == 0 && lom_) fstA[r] = (nb) + (__ffs(lom_) - 1);            \
      if (cntB[r] == 0 && him_) fstB[r] = (nb) + (__ffs(him_) - 1);            \
      cntA[r] += __popc(lom_);                                                 \
      cntB[r] += __popc(him_);                                                 \
    }                                                                          \
  } while (0)

  for (int n0 = 0; n0 < NPTS; n0 += 32) {
    // ---- B matrices: 4x16 (K x N), columns = coords ----
    // Layout: VGPR0 = K0 (lanes 0-15) | K2 (lanes 16-31); VGPR1 = K1 | K3
    const float* cp0 = cb + (size_t)(n0 + c16) * 3;
    const float* cp1 = cp0 + 48;                   // +16 coords
    __builtin_prefetch(cp0 + 96, 0, 3);            // next 32-coord chunk, WGP scope
    const float c0x = cp0[0], c0y = cp0[1], c0z = cp0[2];
    const float c1x = cp1[0], c1y = cp1[1], c1z = cp1[2];

    v2f B0, B1;
    B0.x = lo16 ? c0x : c0z;  B0.y = lo16 ? c0y : 0.0f;
    B1.x = lo16 ? c1x : c1z;  B1.y = lo16 ? c1y : 0.0f;
    const float cn0 = c0x * c0x + c0y * c0y + c0z * c0z;
    const float cn1 = c1x * c1x + c1y * c1y + c1z * c1z;

    v8f acc = {};
    v8f d0 = __builtin_amdgcn_wmma_f32_16x16x4_f32(
        false, A, false, B0, (short)0, acc, false, false);
    v8f d1 = __builtin_amdgcn_wmma_f32_16x16x4_f32(
        false, A, false, B1, (short)0, acc, false, false);

#pragma unroll
    for (int r = 0; r < 8; ++r) PROCESS(d0[r], cn0, n0);
#pragma unroll
    for (int r = 0; r < 8; ++r) PROCESS(d1[r], cn1, n0 + 16);

    // uniform early exit once every row has >= 64 hits
    int mn = cntA[0];
#pragma unroll
    for (int r = 1; r < 8; ++r) mn = min(mn, cntA[r]);
#pragma unroll
    for (int r = 0; r < 8; ++r) mn = min(mn, cntB[r]);
    if (mn >= NS) break;
  }
#undef PROCESS

  // pad positions cnt..63 with first hit (or NPTS when row is empty)
#pragma unroll
  for (int r = 0; r < 8; ++r) {
    int* rp = outBase + (size_t)r * NS;
    if (lane      >= cntA[r]) rp[lane]      = fstA[r];
    if (lane + 32 >= cntA[r]) rp[lane + 32] = fstA[r];
    int* rq = outBase + (size_t)(r + 8) * NS;
    if (lane      >= cntB[r]) rq[lane]      = fstB[r];
    if (lane + 32 >= cntB[r]) rq[lane + 32] = fstB[r];
  }
}

extern "C" void kernel_launch(void* const* d_in, const int* in_sizes, int n_in,
                              void* d_out, int out_size, void* d_ws, size_t ws_size,
                              hipStream_t stream) {
  const float* samples = (const float*)d_in[0];   // (8, 2048, 3) f32
  const float* coord   = (const float*)d_in[1];   // (8, 8192, 3) f32
  int* out = (int*)d_out;                         // (8, 2048, 64) int32

  dim3 grid((BATCH * SPTS / 16) / 8);             // 128 blocks
  dim3 block(256);                                // 8 wave32 per block
  hipLaunchKernelGGL(ball_query_wmma, grid, block, 0, stream, samples, coord, out);
}